// DomainAlign_FaPN_47407849013647
// MI455X (gfx1250) — compile-verified
//
#include <hip/hip_runtime.h>
#include <hip/hip_bf16.h>
#include <math.h>

// ---------------------------------------------------------------------------
// DomainAlign / FaPN deformable-conv pipeline for MI455X (gfx1250, wave32).
//  B=4, C=2, H=2048, W=128, P=128, N=9, KS=3, PAD=1
//  - fm:   pooled avg + 2x2 mix + sigmoid                     (tiny)
//  - mat:  per-(b,c) [2048,128]x[128,128] GEMM  -> f32 WMMA 16x16x4
//  - fs:   2x2 channel mix of mat(+rgb)
//  - deform conv on x=[sar,fs] (gather/VALU bound, L2-resident)
// ---------------------------------------------------------------------------

typedef float v2f __attribute__((ext_vector_type(2)));
typedef float v8f __attribute__((ext_vector_type(8)));

#define HH 2048
#define WW 128
#define HWSZ (HH * WW)

// ------------------------- kernel 1: fm ------------------------------------
__global__ void __launch_bounds__(256) fm_kernel(const float* __restrict__ rgb,
                                                 const float* __restrict__ w_fm,
                                                 float* __restrict__ fm) {
  int idx = blockIdx.x * 256 + threadIdx.x;       // B*128*128 = 65536 threads
  int j = idx & 127;
  int i = (idx >> 7) & 127;
  int b = idx >> 14;
  const float* p0 = rgb + ((size_t)b * 2 + 0) * HWSZ + (size_t)(i * 16) * WW + j;
  const float* p1 = rgb + ((size_t)b * 2 + 1) * HWSZ + (size_t)(i * 16) * WW + j;
  float s0 = 0.f, s1 = 0.f;
#pragma unroll
  for (int r = 0; r < 16; ++r) { s0 += p0[r * WW]; s1 += p1[r * WW]; }
  s0 *= (1.f / 16.f);
  s1 *= (1.f / 16.f);
  float u0 = w_fm[0] * s0 + w_fm[1] * s1;
  float u1 = w_fm[2] * s0 + w_fm[3] * s1;
  fm[((size_t)b * 2 + 0) * 16384 + i * 128 + j] = 1.f / (1.f + __expf(-u0));
  fm[((size_t)b * 2 + 1) * 16384 + i * 128 + j] = 1.f / (1.f + __expf(-u1));
}

// ------------------- kernel 2: mat = rgb@fm + rgb ; fs = w_fs@mat -----------
// Workgroup: 8 waves, one (b, 32-row strip). Wave j owns column tile j (16 cols)
// and both channels; 2 M-tiles x 2 channels = 4 f32 accumulators.
#define APITCH 132   // LDS row pitch (floats), padded to dodge bank conflicts
__global__ void __launch_bounds__(256, 2)
mat_fs_kernel(const float* __restrict__ rgb, const float* __restrict__ fm,
              const float* __restrict__ w_fs, float* __restrict__ fsbuf) {
  __shared__ float aLds[2 * 32 * APITCH];         // [ch][row 0..31][k], ~33 KB

  int b = blockIdx.x >> 6;                        // H/32 = 64 strips per batch
  int strip = blockIdx.x & 63;
  int r0 = strip * 32;
  int tid = threadIdx.x;
  int lane = tid & 31;
  int wv = tid >> 5;                              // wave id = N-tile

  // stage A (rgb rows r0..r0+31, both channels) into LDS, coalesced
  for (int i = tid; i < 8192; i += 256) {
    int c = i >> 12;
    int row = (i >> 7) & 31;
    int k = i & 127;
    aLds[(c * 32 + row) * APITCH + k] =
        rgb[((size_t)b * 2 + c) * HWSZ + (size_t)(r0 + row) * WW + k];
  }
  __syncthreads();

  const float* fm0 = fm + ((size_t)b * 2 + 0) * 16384;
  const float* fm1 = fm + ((size_t)b * 2 + 1) * 16384;

  // WMMA f32 16x16x4 operand placement (wave32):
  //  A: lane l holds row M=l&15, K = 2*(l>>4) + {0,1}       (v2f, contiguous)
  //  B: lane l holds col N=l&15, K = 2*(l>>4) + {0,1}       (row-strided)
  //  C/D: VGPR v holds M = v + 8*(l>>4), N = l&15
  int n16 = wv * 16 + (lane & 15);
  int khalf = (lane >> 4) * 2;
  int mrow = lane & 15;

  v8f acc00 = {}, acc01 = {}, acc10 = {}, acc11 = {};  // [channel][M-tile]
  for (int k0 = 0; k0 < 128; k0 += 4) {
    int kb = k0 + khalf;
    v2f b0, b1;
    b0.x = fm0[kb * 128 + n16];
    b0.y = fm0[(kb + 1) * 128 + n16];
    b1.x = fm1[kb * 128 + n16];
    b1.y = fm1[(kb + 1) * 128 + n16];
    const float* a00p = &aLds[(0 * 32 + mrow) * APITCH + kb];
    const float* a01p = &aLds[(0 * 32 + 16 + mrow) * APITCH + kb];
    const float* a10p = &aLds[(1 * 32 + mrow) * APITCH + kb];
    const float* a11p = &aLds[(1 * 32 + 16 + mrow) * APITCH + kb];
    v2f a00 = {a00p[0], a00p[1]};
    v2f a01 = {a01p[0], a01p[1]};
    v2f a10 = {a10p[0], a10p[1]};
    v2f a11 = {a11p[0], a11p[1]};
    acc00 = __builtin_amdgcn_wmma_f32_16x16x4_f32(false, a00, false, b0,
                                                  (short)0, acc00, false, false);
    acc01 = __builtin_amdgcn_wmma_f32_16x16x4_f32(false, a01, false, b0,
                                                  (short)0, acc01, false, false);
    acc10 = __builtin_amdgcn_wmma_f32_16x16x4_f32(false, a10, false, b1,
                                                  (short)0, acc10, false, false);
    acc11 = __builtin_amdgcn_wmma_f32_16x16x4_f32(false, a11, false, b1,
                                                  (short)0, acc11, false, false);
  }

  // epilogue: mat = acc + rgb ; fs = w_fs @ mat ; write fs scratch
  float wfs00 = w_fs[0], wfs01 = w_fs[1], wfs10 = w_fs[2], wfs11 = w_fs[3];
  int col = wv * 16 + (lane & 15);
  int mofs = (lane >> 4) * 8;
#pragma unroll
  for (int s = 0; s < 2; ++s) {
    v8f ac0 = s ? acc01 : acc00;
    v8f ac1 = s ? acc11 : acc10;
#pragma unroll
    for (int v = 0; v < 8; ++v) {
      int row = r0 + s * 16 + mofs + v;
      float rv0 = rgb[((size_t)b * 2 + 0) * HWSZ + (size_t)row * WW + col];
      float rv1 = rgb[((size_t)b * 2 + 1) * HWSZ + (size_t)row * WW + col];
      float m0 = ac0[v] + rv0;
      float m1 = ac1[v] + rv1;
      fsbuf[((size_t)b * 2 + 0) * HWSZ + (size_t)row * WW + col] =
          wfs00 * m0 + wfs01 * m1;
      fsbuf[((size_t)b * 2 + 1) * HWSZ + (size_t)row * WW + col] =
          wfs10 * m0 + wfs11 * m1;
    }
  }
}

// -------- kernel 3: fused offset conv + bilinear sample + contraction -------
__global__ void __launch_bounds__(256)
deform_kernel(const float* __restrict__ sar, const float* __restrict__ fsbuf,
              const float* __restrict__ w_p, const float* __restrict__ b_p,
              const float* __restrict__ w_dc, float* __restrict__ out) {
  __shared__ float wp[648];   // [18][4][3][3]
  __shared__ float bp[18];
  __shared__ float wdc[72];   // [2][4][9]
  int tid = threadIdx.x;
  for (int i = tid; i < 648; i += 256) wp[i] = w_p[i];
  if (tid < 18) bp[tid] = b_p[tid];
  if (tid < 72) wdc[tid] = w_dc[tid];
  __syncthreads();

  size_t p = (size_t)blockIdx.x * 256 + tid;      // B*H*W threads
  int w = (int)(p & 127);
  int h = (int)((p >> 7) & 2047);
  int b = (int)(p >> 18);

  const float* ch[4];
  ch[0] = sar + ((size_t)b * 2 + 0) * HWSZ;
  ch[1] = sar + ((size_t)b * 2 + 1) * HWSZ;
  ch[2] = fsbuf + ((size_t)b * 2 + 0) * HWSZ;
  ch[3] = fsbuf + ((size_t)b * 2 + 1) * HWSZ;

  // 3x3 neighborhood of x (zero padded), 4 channels
  float xv[4][9];
#pragma unroll
  for (int c = 0; c < 4; ++c) {
#pragma unroll
    for (int ky = 0; ky < 3; ++ky) {
      int y = h + ky - 1;
      bool yok = (unsigned)y < (unsigned)HH;
#pragma unroll
      for (int kx = 0; kx < 3; ++kx) {
        int x = w + kx - 1;
        bool ok = yok && ((unsigned)x < (unsigned)WW);
        xv[c][ky * 3 + kx] = ok ? ch[c][(size_t)y * WW + x] : 0.f;
      }
    }
  }

  // offset conv: 18 outputs, K = 36
  float off[18];
#pragma unroll
  for (int o = 0; o < 18; ++o) {
    float acc = bp[o];
    const float* wpo = &wp[o * 36];
#pragma unroll
    for (int c = 0; c < 4; ++c)
#pragma unroll
      for (int t = 0; t < 9; ++t) acc = fmaf(wpo[c * 9 + t], xv[c][t], acc);
    off[o] = acc;
  }

  const float HpM1 = (float)(HH + 1);   // Hp-1
  const float WpM1 = (float)(WW + 1);   // Wp-1
  float out0 = 0.f, out1 = 0.f;
  for (int n = 0; n < 9; ++n) {
    float py = (float)(h + 1 + (n / 3) - 1) + off[n];
    float px = (float)(w + 1 + (n % 3) - 1) + off[9 + n];
    float fy = floorf(py), fx = floorf(px);
    float y0f = fminf(fmaxf(fy, 0.f), HpM1);
    float y1f = fminf(fmaxf(fy + 1.f, 0.f), HpM1);
    float x0f = fminf(fmaxf(fx, 0.f), WpM1);
    float x1f = fminf(fmaxf(fx + 1.f, 0.f), WpM1);
    float pyc = fminf(fmaxf(py, 0.f), HpM1);
    float pxc = fminf(fmaxf(px, 0.f), WpM1);
    int y0 = (int)y0f, y1 = (int)y1f, x0 = (int)x0f, x1 = (int)x1f;
    float gy0 = 1.f + (y0f - pyc);
    float gy1 = 1.f - (y1f - pyc);
    float gx0 = 1.f + (x0f - pxc);
    float gx1 = 1.f - (x1f - pxc);
    float g_lt = gy0 * gx0, g_rb = gy1 * gx1, g_lb = gy0 * gx1, g_rt = gy1 * gx0;
    // padded coords -> unpadded image; pad ring samples zero
    int yi0 = y0 - 1, yi1 = y1 - 1, xi0 = x0 - 1, xi1 = x1 - 1;
    bool vy0 = (unsigned)yi0 < (unsigned)HH;
    bool vy1 = (unsigned)yi1 < (unsigned)HH;
    bool vx0 = (unsigned)xi0 < (unsigned)WW;
    bool vx1 = (unsigned)xi1 < (unsigned)WW;
#pragma unroll
    for (int c = 0; c < 4; ++c) {
      const float* xc = ch[c];
      float v00 = (vy0 && vx0) ? xc[(size_t)yi0 * WW + xi0] : 0.f;
      float v11 = (vy1 && vx1) ? xc[(size_t)yi1 * WW + xi1] : 0.f;
      float v01 = (vy0 && vx1) ? xc[(size_t)yi0 * WW + xi1] : 0.f;
      float v10 = (vy1 && vx0) ? xc[(size_t)yi1 * WW + xi0] : 0.f;
      float val = g_lt * v00 + g_rb * v11 + g_lb * v01 + g_rt * v10;
      out0 = fmaf(wdc[0 * 36 + c * 9 + n], val, out0);
      out1 = fmaf(wdc[1 * 36 + c * 9 + n], val, out1);
    }
  }
  out[((size_t)b * 2 + 0) * HWSZ + (size_t)h * WW + w] = out0;
  out[((size_t)b * 2 + 1) * HWSZ + (size_t)h * WW + w] = out1;
}

// ---------------------------------------------------------------------------
extern "C" void kernel_launch(void* const* d_in, const int* in_sizes, int n_in,
                              void* d_out, int out_size, void* d_ws, size_t ws_size,
                              hipStream_t stream) {
  const float* rgb  = (const float*)d_in[0];
  const float* sar  = (const float*)d_in[1];
  const float* w_fm = (const float*)d_in[2];
  const float* w_fs = (const float*)d_in[3];
  const float* w_p  = (const float*)d_in[4];
  const float* b_p  = (const float*)d_in[5];
  const float* w_dc = (const float*)d_in[6];
  float* out = (float*)d_out;

  // scratch layout: fm [B,2,128,128] then fs [B,2,H,W]  (~8.9 MB total)
  float* fm    = (float*)d_ws;
  float* fsbuf = fm + 4 * 2 * 128 * 128;

  fm_kernel<<<256, 256, 0, stream>>>(rgb, w_fm, fm);
  mat_fs_kernel<<<256, 256, 0, stream>>>(rgb, fm, w_fs, fsbuf);
  deform_kernel<<<4096, 256, 0, stream>>>(sar, fsbuf, w_p, b_p, w_dc, out);
}